// VectorQuantizer_78159814852716
// MI455X (gfx1250) — compile-verified
//
#include <hip/hip_runtime.h>
#include <stdint.h>

typedef __attribute__((ext_vector_type(2))) float v2f;
typedef __attribute__((ext_vector_type(4))) float v4f;
typedef __attribute__((ext_vector_type(8))) float v8f;

#define K_CODES   512
#define D_DIM     64
#define HW        4096          // H*W
#define LDS_STR   68            // padded row stride in floats (272B = 17*16B, conflict-free b64 reads)
#define MT        2             // row-tiles (of 16 rows) per wave

static __device__ __forceinline__ v2f mkv2(float a, float b) { v2f r; r.x = a; r.y = b; return r; }

__global__ __launch_bounds__(256) void vq_wmma_kernel(const float* __restrict__ x,
                                                      const float* __restrict__ emb,
                                                      float* __restrict__ out) {
  extern __shared__ float smem[];
  float* s_emb = smem;                        // [512][LDS_STR]
  float* s_esq = smem + K_CODES * LDS_STR;    // [512]

  const int tid  = threadIdx.x;
  const int lane = tid & 31;
  const int wave = tid >> 5;
  const int l16  = lane & 15;
  const int half = lane >> 4;

  // ---- stage codebook into LDS via async DMA (global -> LDS, no VGPR round trip) ----
  // 128KB = 8192 x 16B chunks; 256 threads x 32 chunks. LDS layout keeps the padded
  // stride (68 floats/row): chunk c -> row c/16, 16B-slot c%16.
  {
    const uint64_t gbase    = (uint64_t)(uintptr_t)emb;
    const uint32_t lds_base = (uint32_t)(uintptr_t)s_emb;  // low 32 bits of generic = LDS byte addr
#pragma unroll
    for (int i = 0; i < 32; ++i) {
      const int chunk = tid + i * 256;                 // 0..8191
      const int row   = chunk >> 4;
      const int c16   = chunk & 15;
      const uint32_t goff = (uint32_t)chunk * 16u;                       // global byte offset
      const uint32_t loff = lds_base + (uint32_t)row * (LDS_STR * 4) + (uint32_t)c16 * 16u;
      asm volatile("global_load_async_to_lds_b128 %0, %1, %2"
                   :: "v"(loff), "v"(goff), "s"(gbase)
                   : "memory");
    }
  }

  // ---- overlap: e_sq from global (emb is hot in L2) while the DMA runs ----
  for (int k = tid; k < K_CODES; k += 256) {
    const float* src = emb + k * D_DIM;
    float sum = 0.f;
#pragma unroll
    for (int d = 0; d < D_DIM; d += 4) {
      v4f v = *(const v4f*)(src + d);
      sum += v.x * v.x + v.y * v.y + v.z * v.z + v.w * v.w;
    }
    s_esq[k] = sum;
  }

  asm volatile("s_wait_asynccnt 0x0" ::: "memory");   // this wave's DMAs landed in LDS
  __syncthreads();                                    // all waves' DMAs + e_sq visible

  // ---- load A fragments: x tile rows, WMMA f32 16x16x4 A layout ----
  // lane L: row = l16, VGPR v holds K = v + 2*half within each 4-wide d-chunk
  const int n0 = blockIdx.x * (8 * MT * 16) + wave * (MT * 16);
  v2f afrag[MT][16];
#pragma unroll
  for (int t = 0; t < MT; ++t) {
    const int row = n0 + t * 16 + l16;            // global pixel row (b,h,w)
    const int b   = row >> 12;                    // row / HW
    const int hw  = row & (HW - 1);
    const float* xb = x + (size_t)b * D_DIM * HW + hw;
#pragma unroll
    for (int j = 0; j < 16; ++j) {
      const int c0 = 4 * j + 2 * half;
      afrag[t][j] = mkv2(xb[(size_t)c0 * HW], xb[(size_t)(c0 + 1) * HW]);
    }
  }

  float bestS[MT][8];
  int   bestK[MT][8];
#pragma unroll
  for (int t = 0; t < MT; ++t)
#pragma unroll
    for (int r = 0; r < 8; ++r) { bestS[t][r] = 3.4e38f; bestK[t][r] = 0; }

  // ---- k-tile loop: cross = x . e via WMMA, fold into running argmin ----
  for (int kt = 0; kt < K_CODES; kt += 16) {
    v8f acc[MT];
#pragma unroll
    for (int t = 0; t < MT; ++t) acc[t] = v8f{0.f, 0.f, 0.f, 0.f, 0.f, 0.f, 0.f, 0.f};

    const float* brow = s_emb + (kt + l16) * LDS_STR + 2 * half;  // B: N=l16, K=v+2*half
#pragma unroll
    for (int j = 0; j < 16; ++j) {
      v2f bf = *(const v2f*)(brow + 4 * j);
#pragma unroll
      for (int t = 0; t < MT; ++t)
        acc[t] = __builtin_amdgcn_wmma_f32_16x16x4_f32(
            false, afrag[t][j], false, bf, (short)0, acc[t], false, false);
    }

    const float esq = s_esq[kt + l16];
#pragma unroll
    for (int t = 0; t < MT; ++t)
#pragma unroll
      for (int r = 0; r < 8; ++r) {
        // dist = x_sq + e_sq - 2*cross; x_sq is constant per row -> drop it
        float s = esq - 2.0f * acc[t][r];
        if (s < bestS[t][r]) { bestS[t][r] = s; bestK[t][r] = kt; }  // strict <: first min wins
      }
  }

  // ---- reduce argmin across the 16 lanes sharing each row ----
  int bestI[MT][8];
#pragma unroll
  for (int t = 0; t < MT; ++t)
#pragma unroll
    for (int r = 0; r < 8; ++r) {
      float s = bestS[t][r];
      int   i = bestK[t][r] + l16;                 // column N = l16
#pragma unroll
      for (int m = 1; m <= 8; m <<= 1) {           // xor masks stay within 16-lane halves
        float os = __shfl_xor(s, m, 32);
        int   oi = __shfl_xor(i, m, 32);
        if (os < s || (os == s && oi < i)) { s = os; i = oi; }
      }
      bestI[t][r] = i;
    }

  // ---- gather winning code rows from LDS, coalesced 256B/row store ----
#pragma unroll
  for (int t = 0; t < MT; ++t) {
    const int rowbase = n0 + t * 16;
#pragma unroll
    for (int m = 0; m < 16; ++m) {
      // row M=m lives in half (m<8?lo:hi), register r=m&7; all lanes of that half agree
      int idx = __shfl(bestI[t][m & 7], (m < 8) ? 0 : 16, 32);
      v2f v = *(const v2f*)(s_emb + idx * LDS_STR + 2 * lane);
      *(v2f*)(out + (size_t)(rowbase + m) * D_DIM + 2 * lane) = v;
    }
  }
}

extern "C" void kernel_launch(void* const* d_in, const int* in_sizes, int n_in,
                              void* d_out, int out_size, void* d_ws, size_t ws_size,
                              hipStream_t stream) {
  const float* x   = (const float*)d_in[0];   // [32,64,64,64] f32
  const float* emb = (const float*)d_in[1];   // [512,64] f32
  float* out = (float*)d_out;                 // [32,64,64,64] f32 (flat [N,64] rows)

  const int N = 32 * HW;                      // 131072 pixel rows
  const int rows_per_block = 8 * MT * 16;     // 256
  dim3 grid(N / rows_per_block), block(256);
  size_t shmem = (size_t)(K_CODES * LDS_STR + K_CODES) * sizeof(float); // ~138 KB < 320 KB/WGP
  hipLaunchKernelGGL(vq_wmma_kernel, grid, block, shmem, stream, x, emb, out);
}